// AdaptiveRankLinear_75041668596068
// MI455X (gfx1250) — compile-verified
//
#include <hip/hip_runtime.h>

typedef __attribute__((ext_vector_type(2))) float v2f;
typedef __attribute__((ext_vector_type(4))) float v4f;
typedef __attribute__((ext_vector_type(8))) float v8f;

#define DIM   4096
#define RANK  16
#define ROWS_PER_WG 16
#define WAVES 8

__global__ __launch_bounds__(256) void ar_fused_kernel(
    const float* __restrict__ x, const float* __restrict__ U,
    const float* __restrict__ S, const float* __restrict__ V,
    const float* __restrict__ nw, const float* __restrict__ gam,
    float* __restrict__ out)
{
  __shared__ float red[256];        // sumsq reduction
  __shared__ float rms[16];
  __shared__ float smask[16];       // S * keep
  __shared__ float hpart[WAVES][256];
  __shared__ float H[256];          // final 16x16 H, row-major [m][r]

  const int t    = threadIdx.x;
  const int wave = t >> 5;
  const int lane = t & 31;
  const long rowBase = (long)blockIdx.x * ROWS_PER_WG;
  const float* xt = x + rowBase * DIM;

  // ---- rank mask (tiny, redundant per block) ----
  if (t == 0) {
    float tot = 0.f;
    for (int i = 0; i < RANK; ++i) tot += fabsf(S[i]);
    tot = fmaxf(tot, 1e-8f);
    float cum = 0.f; int r = RANK;
    for (int i = 0; i < RANK; ++i) {
      cum += fabsf(S[i]);
      if (cum >= 0.95f * tot) { r = i + 1; break; }
    }
    for (int i = 0; i < RANK; ++i) smask[i] = (i < r) ? S[i] : 0.f;
  }

  // ---- Phase A: per-row sum of squares (16 threads per row) ----
  {
    const int row = t >> 4;
    const int lr  = t & 15;
    const float* xr = xt + row * DIM;
    float ss = 0.f;
    for (int c = lr * 4; c < DIM; c += 64) {
      v4f v = *(const v4f*)(xr + c);
      ss += v.x * v.x + v.y * v.y + v.z * v.z + v.w * v.w;
    }
    red[row * 16 + lr] = ss;
  }
  __syncthreads();
  if (t < 16) {
    float s = 0.f;
    for (int j = 0; j < 16; ++j) s += red[t * 16 + j];
    rms[t] = rsqrtf(s * (1.0f / DIM) + 1e-6f);
  }
  __syncthreads();

  // ---- Phase B: H = (x*rms*w) @ U  via v_wmma_f32_16x16x4_f32, K split over waves ----
  {
    const int mrow = lane & 15;            // A row / B col index
    const int koff = (lane >> 4) << 1;     // 0 (lanes 0-15) or 2 (lanes 16-31)
    const float* xr = xt + mrow * DIM;
    const float rm = rms[mrow];
    v8f c = {};
    const int kb = wave * (DIM / WAVES);
    const int ke = kb + (DIM / WAVES);
    for (int k0 = kb; k0 < ke; k0 += 4) {
      const int k = k0 + koff;
      v2f a, b;
      a.x = xr[k]     * rm * nw[k];
      a.y = xr[k + 1] * rm * nw[k + 1];
      b.x = U[k * RANK + mrow];
      b.y = U[(k + 1) * RANK + mrow];
      c = __builtin_amdgcn_wmma_f32_16x16x4_f32(false, a, false, b,
                                                (short)0, c, false, false);
    }
    const int mbase = (lane >> 4) * 8;
#pragma unroll
    for (int v = 0; v < 8; ++v)
      hpart[wave][(mbase + v) * 16 + mrow] = c[v];
  }
  __syncthreads();
  {
    float s = 0.f;
#pragma unroll
    for (int w = 0; w < WAVES; ++w) s += hpart[w][t];
    H[t] = s * smask[t & 15];    // N index = rank
  }
  __syncthreads();

  // ---- Phase C: out = x + (H @ V) * gamma, N split over waves ----
  {
    const int mrow = lane & 15;
    const int ncol = lane & 15;
    const int koff = (lane >> 4) << 1;
    v2f afr[4];
#pragma unroll
    for (int sI = 0; sI < 4; ++sI) {
      const int k = 4 * sI + koff;
      afr[sI].x = H[mrow * 16 + k];
      afr[sI].y = H[mrow * 16 + k + 1];
    }
    const int nb = wave * (DIM / WAVES);
    const int mbase = (lane >> 4) * 8;
    for (int n0 = nb; n0 < nb + (DIM / WAVES); n0 += 16) {
      v8f d = {};
#pragma unroll
      for (int sI = 0; sI < 4; ++sI) {
        const int k = 4 * sI + koff;
        v2f b;
        b.x = V[(long)k * DIM + n0 + ncol];
        b.y = V[(long)(k + 1) * DIM + n0 + ncol];
        d = __builtin_amdgcn_wmma_f32_16x16x4_f32(false, afr[sI], false, b,
                                                  (short)0, d, false, false);
      }
      const float g = gam[n0 + ncol];
#pragma unroll
      for (int v = 0; v < 8; ++v) {
        const long idx = (rowBase + mbase + v) * (long)DIM + n0 + ncol;
        out[idx] = x[idx] + d[v] * g;
      }
    }
  }
}

extern "C" void kernel_launch(void* const* d_in, const int* in_sizes, int n_in,
                              void* d_out, int out_size, void* d_ws, size_t ws_size,
                              hipStream_t stream) {
  const float* x   = (const float*)d_in[0];
  const float* U   = (const float*)d_in[1];
  const float* S   = (const float*)d_in[2];
  const float* V   = (const float*)d_in[3];
  const float* nw  = (const float*)d_in[4];
  const float* gam = (const float*)d_in[5];
  float* out = (float*)d_out;

  const int rows = in_sizes[0] / DIM;          // 8192
  const int grid = rows / ROWS_PER_WG;         // 512
  ar_fused_kernel<<<grid, 256, 0, stream>>>(x, U, S, V, nw, gam, out);
}